// PEAGATChannel_54812372631692
// MI455X (gfx1250) — compile-verified
//
#include <hip/hip_runtime.h>
#include <math.h>

// ---------------------------------------------------------------------------
// 2-layer GAT for gfx1250 (MI455X).  GEMMs use v_wmma_f32_16x16x4_f32 (full
// fp32 precision; GEMMs are not the bottleneck).  Edge softmax/scatter uses
// f32 atomics which resolve in the 192MB L2 (h1 = 51MB fits entirely).
// ---------------------------------------------------------------------------

typedef float v2f __attribute__((ext_vector_type(2)));
typedef float v8f __attribute__((ext_vector_type(8)));

#define NN      50000
#define NE      800000
#define EMB     128
#define H1      4
#define DD      64      // per-head dim for both layers
#define C1      256     // H1*DD
#define C2      64
#define NEG_SLOPE 0.2f

// ---------------- WMMA fp32 GEMM:  C[M,N] = A[M,K] @ B[K,N] -----------------
// Each wave computes a 16(M) x 64(N) strip (4 accumulators), reusing one A
// fragment per 4 WMMAs.  4 waves per block => block covers 64(M) x 64(N).
// REQUIRES: M % 16 == 0, N % 64 == 0, K % 4 == 0 (true for all our launches:
// M=50000=3125*16, N in {256,64}, K in {128,256}).
// ISA layouts (CDNA5 ISA):
//   A 16x4:  lanes 0-15 rows M=0..15; VGPR0 = K=(0|2), VGPR1 = K=(1|3)
//   B 4x16:  lanes 0-15 cols N=0..15; VGPR0 = K=(0|2), VGPR1 = K=(1|3)
//   C 16x16: VGPR i: lanes 0-15 -> M=i, lanes 16-31 -> M=8+i
__global__ __launch_bounds__(128)
void wmma_gemm_f32(const float* __restrict__ A, const float* __restrict__ B,
                   float* __restrict__ C, int M, int N, int K) {
    const int wave  = threadIdx.x >> 5;
    const int lane  = threadIdx.x & 31;
    const int mTile = blockIdx.x * 4 + wave;
    if (mTile * 16 >= M) return;            // wave-uniform exit

    const int l15  = lane & 15;
    const int hi   = lane >> 4;             // 0 or 1 (half-wave -> K pair)
    const int row  = mTile * 16 + l15;      // A row for this lane
    const int col0 = blockIdx.y * 64 + l15; // first of 4 C columns

    // Running pointers: compiler strength-reduces to constant-offset loads.
    const v2f*   __restrict__ ap = (const v2f*)(A + (size_t)row * K + hi * 2);
    const float* __restrict__ bp = B + (size_t)(hi * 2) * N + col0;

    v8f acc0 = {}, acc1 = {}, acc2 = {}, acc3 = {};
    #pragma unroll 2
    for (int k0 = 0; k0 < K; k0 += 4) {
        const v2f a = ap[0]; ap += 2;                 // global_load_b64
        v2f b0, b1, b2, b3;
        b0.x = bp[0];  b0.y = bp[N];
        b1.x = bp[16]; b1.y = bp[N + 16];
        b2.x = bp[32]; b2.y = bp[N + 32];
        b3.x = bp[48]; b3.y = bp[N + 48];
        bp += (size_t)4 * N;
        acc0 = __builtin_amdgcn_wmma_f32_16x16x4_f32(false, a, false, b0, (short)0, acc0, false, false);
        acc1 = __builtin_amdgcn_wmma_f32_16x16x4_f32(false, a, false, b1, (short)0, acc1, false, false);
        acc2 = __builtin_amdgcn_wmma_f32_16x16x4_f32(false, a, false, b2, (short)0, acc2, false, false);
        acc3 = __builtin_amdgcn_wmma_f32_16x16x4_f32(false, a, false, b3, (short)0, acc3, false, false);
    }

    // Store: VGPR i holds row rbase+i for this lane's column.
    const int rbase = mTile * 16 + hi * 8;
    float* __restrict__ cp = C + (size_t)rbase * N + col0;
    #pragma unroll
    for (int i = 0; i < 8; ++i) {
        cp[0]  = acc0[i];
        cp[16] = acc1[i];
        cp[32] = acc2[i];
        cp[48] = acc3[i];
        cp += N;
    }
}

// -------- per-node attention dots: a_s[n,h]=<h[n,h,:],att_s[h,:]> -----------
__global__ void att_dots(const float* __restrict__ h,
                         const float* __restrict__ att_s,
                         const float* __restrict__ att_d,
                         float* __restrict__ as_, float* __restrict__ ad_,
                         int N, int H) {
    int t = blockIdx.x * blockDim.x + threadIdx.x;
    if (t >= N * H) return;
    int n = t / H, hh = t % H;
    const float* hp = h + (size_t)n * H * DD + hh * DD;
    const float* sp = att_s + hh * DD;
    const float* dp = att_d + hh * DD;
    float s = 0.f, d = 0.f;
    #pragma unroll 8
    for (int i = 0; i < DD; ++i) { s += hp[i] * sp[i]; d += hp[i] * dp[i]; }
    as_[t] = s; ad_[t] = d;
}

__global__ void fill_f32(float* __restrict__ p, float v, int n) {
    int t = blockIdx.x * blockDim.x + threadIdx.x;
    if (t < n) p[t] = v;
}

__device__ __forceinline__ float leaky(float x) {
    return x > 0.f ? x : NEG_SLOPE * x;
}

// Order-preserving float atomic max via int/uint monotone bit tricks.
__device__ __forceinline__ void atomicMaxF(float* addr, float v) {
    if (v >= 0.f) atomicMax((int*)addr, __float_as_int(v));
    else          atomicMin((unsigned int*)addr, __float_as_uint(v));
}

__device__ __forceinline__ void edge_ids(const int* __restrict__ ei, int e, int E,
                                         int& src, int& dst) {
    if (e < E) { src = ei[e]; dst = ei[E + e]; }   // edge_index[0]=src, [1]=dst
    else       { src = dst = e - E; }              // self-loop
}

// ---- pass 1: segment max of leaky_relu(a_s[src]+a_d[dst]) over dst ---------
__global__ void edge_max(const int* __restrict__ ei, int E, int N, int H,
                         const float* __restrict__ as_, const float* __restrict__ ad_,
                         float* __restrict__ m) {
    int t = blockIdx.x * blockDim.x + threadIdx.x;
    if (t >= (E + N) * H) return;
    int e = t / H, hh = t % H, src, dst;
    edge_ids(ei, e, E, src, dst);
    float v = leaky(as_[src * H + hh] + ad_[dst * H + hh]);
    atomicMaxF(&m[dst * H + hh], v);
}

// ---- pass 2: denom[dst] += exp(e - m[dst]) ---------------------------------
__global__ void edge_denom(const int* __restrict__ ei, int E, int N, int H,
                           const float* __restrict__ as_, const float* __restrict__ ad_,
                           const float* __restrict__ m, float* __restrict__ denom) {
    int t = blockIdx.x * blockDim.x + threadIdx.x;
    if (t >= (E + N) * H) return;
    int e = t / H, hh = t % H, src, dst;
    edge_ids(ei, e, E, src, dst);
    int di = dst * H + hh;
    float v = leaky(as_[src * H + hh] + ad_[di]);
    atomicAdd(&denom[di], __expf(v - m[di]));
}

// ---- pass 3: acc[dst,h,:] += alpha * h[src,h,:]  (64 lanes per (edge,head)) -
__global__ __launch_bounds__(256)
void edge_scatter(const int* __restrict__ ei, int E, int N, int H,
                  const float* __restrict__ as_, const float* __restrict__ ad_,
                  const float* __restrict__ m, const float* __restrict__ denom,
                  const float* __restrict__ h, float* __restrict__ acc) {
    int unit = blockIdx.x * 4 + (threadIdx.x >> 6);   // (edge, head) unit
    int d    = threadIdx.x & 63;
    if (unit >= (E + N) * H) return;
    int e = unit / H, hh = unit % H, src, dst;
    edge_ids(ei, e, E, src, dst);
    int di = dst * H + hh;
    float v     = leaky(as_[src * H + hh] + ad_[di]);
    float alpha = __expf(v - m[di]) / denom[di];
    atomicAdd(&acc[(size_t)dst * H * DD + hh * DD + d],
              alpha * h[(size_t)src * H * DD + hh * DD + d]);
}

// ---- out = (acc + bias) [relu] ---------------------------------------------
__global__ void bias_act(const float* __restrict__ acc, const float* __restrict__ bias,
                         float* __restrict__ out, int N, int C, int doRelu) {
    int t = blockIdx.x * blockDim.x + threadIdx.x;
    if (t >= N * C) return;
    float v = acc[t] + bias[t % C];
    out[t] = doRelu ? fmaxf(v, 0.f) : v;
}

// ---------------------------------------------------------------------------
extern "C" void kernel_launch(void* const* d_in, const int* in_sizes, int n_in,
                              void* d_out, int out_size, void* d_ws, size_t ws_size,
                              hipStream_t stream) {
    const float* x        = (const float*)d_in[0];
    const float* W1       = (const float*)d_in[1];
    const float* att_src1 = (const float*)d_in[2];
    const float* att_dst1 = (const float*)d_in[3];
    const float* b1       = (const float*)d_in[4];
    const float* W2       = (const float*)d_in[5];
    const float* att_src2 = (const float*)d_in[6];
    const float* att_dst2 = (const float*)d_in[7];
    const float* b2       = (const float*)d_in[8];
    const int*   ei0      = (const int*)d_in[9];
    const int*   ei1      = (const int*)d_in[10];
    float*       out      = (float*)d_out;

    // workspace layout (floats)
    float* ws = (float*)d_ws;
    size_t o = 0;
    float* h1     = ws + o; o += (size_t)NN * C1;   // GEMM1 out, later relu'd layer-2 input
    float* acc1   = ws + o; o += (size_t)NN * C1;   // layer-1 message accumulator
    float* h2     = ws + o; o += (size_t)NN * C2;   // GEMM2 out
    float* as1    = ws + o; o += (size_t)NN * H1;
    float* ad1    = ws + o; o += (size_t)NN * H1;
    float* m1     = ws + o; o += (size_t)NN * H1;
    float* dn1    = ws + o; o += (size_t)NN * H1;
    float* as2    = ws + o; o += (size_t)NN;
    float* ad2    = ws + o; o += (size_t)NN;
    float* m2     = ws + o; o += (size_t)NN;
    float* dn2    = ws + o; o += (size_t)NN;

    const int B = 256;
    const int mTiles = (NN + 15) / 16;
    const int EH1 = (NE + NN) * H1;
    const int EH2 = (NE + NN) * 1;

    // ---------------- layer 1 ----------------
    {
        dim3 grid((mTiles + 3) / 4, C1 / 64);
        wmma_gemm_f32<<<grid, 128, 0, stream>>>(x, W1, h1, NN, C1, EMB);
    }
    att_dots<<<(NN * H1 + B - 1) / B, B, 0, stream>>>(h1, att_src1, att_dst1, as1, ad1, NN, H1);
    fill_f32<<<(NN * H1 + B - 1) / B, B, 0, stream>>>(m1, -INFINITY, NN * H1);
    fill_f32<<<(NN * H1 + B - 1) / B, B, 0, stream>>>(dn1, 0.f, NN * H1);
    fill_f32<<<((size_t)NN * C1 + B - 1) / B, B, 0, stream>>>(acc1, 0.f, NN * C1);
    edge_max   <<<(EH1 + B - 1) / B, B, 0, stream>>>(ei0, NE, NN, H1, as1, ad1, m1);
    edge_denom <<<(EH1 + B - 1) / B, B, 0, stream>>>(ei0, NE, NN, H1, as1, ad1, m1, dn1);
    edge_scatter<<<(EH1 + 3) / 4, 256, 0, stream>>>(ei0, NE, NN, H1, as1, ad1, m1, dn1, h1, acc1);
    // h1 <- relu(acc1 + b1)   (h1 no longer needed as messages)
    bias_act<<<((size_t)NN * C1 + B - 1) / B, B, 0, stream>>>(acc1, b1, h1, NN, C1, 1);

    // ---------------- layer 2 ----------------
    {
        dim3 grid((mTiles + 3) / 4, C2 / 64);
        wmma_gemm_f32<<<grid, 128, 0, stream>>>(h1, W2, h2, NN, C2, C1);
    }
    att_dots<<<(NN + B - 1) / B, B, 0, stream>>>(h2, att_src2, att_dst2, as2, ad2, NN, 1);
    fill_f32<<<(NN + B - 1) / B, B, 0, stream>>>(m2, -INFINITY, NN);
    fill_f32<<<(NN + B - 1) / B, B, 0, stream>>>(dn2, 0.f, NN);
    fill_f32<<<((size_t)NN * C2 + B - 1) / B, B, 0, stream>>>(out, 0.f, NN * C2);
    edge_max   <<<(EH2 + B - 1) / B, B, 0, stream>>>(ei1, NE, NN, 1, as2, ad2, m2);
    edge_denom <<<(EH2 + B - 1) / B, B, 0, stream>>>(ei1, NE, NN, 1, as2, ad2, m2, dn2);
    edge_scatter<<<(EH2 + 3) / 4, 256, 0, stream>>>(ei1, NE, NN, 1, as2, ad2, m2, dn2, h2, out);
    // out += b2 (in place, no relu)
    bias_act<<<((size_t)NN * C2 + B - 1) / B, B, 0, stream>>>(out, b2, out, NN, C2, 0);
}